// Linear_dwa_42468636622849
// MI455X (gfx1250) — compile-verified
//
#include <hip/hip_runtime.h>

// Problem constants (from reference)
#define FIN   1024
#define FOUT  1024
#define BATCH 256

// Tiling
#define BT 2                    // batch rows per block
#define OTILE 32                // output columns per block (2 groups of 16)
#define FS 64                   // f-slice length
#define NSLICE (FIN / FS)       // 16 slices
#define ROWF (FS + 4)           // padded LDS row stride in floats (68 % 64 == 4 -> conflict-free)
#define MROWS (BT * OTILE)      // 64 rows in m slab
#define WROWS (OTILE)           // 32 rows in w slab

typedef unsigned int       u32;
typedef unsigned long long u64;
typedef __attribute__((ext_vector_type(4))) unsigned int v4u;
typedef __attribute__((ext_vector_type(4))) int          v4i;
typedef __attribute__((ext_vector_type(8))) int          v8i;
typedef __attribute__((ext_vector_type(2))) float        v2f;
typedef __attribute__((ext_vector_type(8))) float        v8f;

// Issue one Tensor-Data-Mover DMA of f32 data global -> LDS with hardware row
// padding (pad 4 DWORDs after every 64 DWORDs => LDS row stride 68 floats).
// tile: [tile_d2][32 rows][FS floats]; row stride FIN, plane stride FOUT*FIN.
__device__ __forceinline__ void tdm_load_tile(const float* gbase, u32 lds_byte_off,
                                              int tile_d2) {
  const u64 ga  = (u64)(size_t)gbase;
  const u32 TD0 = 1u << 20;                 // generous tensor dims (tile is in-bounds)
  const u32 TD1 = 1u << 20;
  const u64 S1  = (u64)FOUT * (u64)FIN;     // dim1 (batch) stride, elements

  v4u g0;
  g0[0] = 1u;                                            // count=1, user descriptor
  g0[1] = lds_byte_off;                                  // lds_addr
  g0[2] = (u32)(ga & 0xFFFFFFFFull);                     // global_addr[31:0]
  g0[3] = ((u32)(ga >> 32) & 0x01FFFFFFu) | (2u << 30);  // global_addr[56:32] | type=2

  u32 d1[8];
  d1[0] = (2u << 16)        // data_size = 4B
        | (1u << 20)        // pad_enable
        | (5u << 22)        // pad_interval code 5 = 64 DWORDs
        | (3u << 25);       // pad_amount  code 3 = 4 DWORDs
  d1[1] = (TD0 & 0xFFFFu) << 16;                               // tensor_dim0 lo16
  d1[2] = ((TD0 >> 16) & 0xFFFFu) | ((TD1 & 0xFFFFu) << 16);   // dim0 hi | dim1 lo
  d1[3] = ((TD1 >> 16) & 0xFFFFu) | ((u32)FS << 16);           // dim1 hi | tile_dim0=FS
  d1[4] = ((u32)tile_d2 << 16) | (u32)OTILE;                   // tile_dim1=32, tile_dim2
  d1[5] = (u32)FIN;                                            // dim0 stride lo32
  d1[6] = 0u;                                                  // stride0 hi | stride1 lo16 (0)
  d1[7] = (u32)(S1 >> 16);                                     // stride1 bits [47:16]
  v8i g1;
#pragma unroll
  for (int k = 0; k < 8; ++k) g1[k] = (int)d1[k];

  v4i g2 = {(tile_d2 > 0) ? (int)(1u << 20) : 0, 0, 0, 0};     // tensor_dim2
  v4i g3 = {0, 0, 0, 0};

#if __clang_major__ >= 23
  v8i g4 = {0, 0, 0, 0, 0, 0, 0, 0};
  __builtin_amdgcn_tensor_load_to_lds(g0, g1, g2, g3, g4, 0);
#else
  __builtin_amdgcn_tensor_load_to_lds(g0, g1, g2, g3, 0);
#endif
}

__global__ __launch_bounds__(128) void fused_dwa_wmma_kernel(
    const float* __restrict__ x,      // [BATCH][FIN]
    const float* __restrict__ m,      // [BATCH][FOUT][FIN]  (1 GiB stream, bandwidth bound)
    const float* __restrict__ w,      // [FOUT][FIN]
    const float* __restrict__ bias,   // [FOUT]
    float* __restrict__ out) {        // [BATCH][FOUT]
  __shared__ float xs[BT * FIN];                 // 8 KB  : x rows (unpadded, broadcast reads)
  __shared__ float msh[2][MROWS * ROWF];         // 34 KB : TDM double buffer, m (padded rows)
  __shared__ float wsh[2][WROWS * ROWF];         // 17 KB : TDM double buffer, w (padded rows)

  const int tid  = threadIdx.x;
  const int wid  = tid >> 5;            // wave id, 0..3 (wave32)
  const int lane = tid & 31;
  const int h    = lane >> 4;           // lane half: selects K offset 2*h
  const int r    = lane & 15;           // matrix row within 16-o group

  const int btile = blockIdx.x / (FOUT / OTILE);   // 0..127
  const int otile = blockIdx.x % (FOUT / OTILE);   // 0..31
  const int b0 = btile * BT;
  const int o0 = otile * OTILE;
  const int bi = wid >> 1;              // this wave's batch row (0..1)
  const int g  = wid & 1;               // this wave's 16-o group (0..1)

  // Stage x rows [b0, b0+BT) into LDS once (coalesced float4 loads).
#pragma unroll
  for (int i = 0; i < (BT * FIN / 4) / 128; ++i) {
    int c  = tid + i * 128;                       // 0..511 float4 chunks
    int rr = c / (FIN / 4);
    int k  = c % (FIN / 4);
    ((float4*)xs)[c] = ((const float4*)(x + (size_t)(b0 + rr) * FIN))[k];
  }

  const float* mtile = m + ((size_t)b0 * FOUT + (size_t)o0) * (size_t)FIN;
  const float* wtile = w + (size_t)o0 * (size_t)FIN;

  // Kick off slice 0 DMAs (wave 0 only; TDM ops are wave-level, tracked by TENSORcnt).
  if (wid == 0) {
    tdm_load_tile(mtile, (u32)(size_t)&msh[0][0], BT);
    tdm_load_tile(wtile, (u32)(size_t)&wsh[0][0], 0);
  }

  // Per-lane LDS base offsets (floats) for the WMMA A-operand gather:
  // lane l supplies A rows M=r at K = vgpr + 2*h  ->  float2 at (row, 2*h).
  const int mrow_off = (bi * OTILE + g * 16 + r) * ROWF + 2 * h;
  const int wrow_off = (g * 16 + r) * ROWF + 2 * h;

  v8f c_acc;
#pragma unroll
  for (int k = 0; k < 8; ++k) c_acc[k] = 0.0f;

  for (int s = 0; s < NSLICE; ++s) {
    if (wid == 0) {
      if (s + 1 < NSLICE) {
        tdm_load_tile(mtile + (size_t)(s + 1) * FS, (u32)(size_t)&msh[(s + 1) & 1][0], BT);
        tdm_load_tile(wtile + (size_t)(s + 1) * FS, (u32)(size_t)&wsh[(s + 1) & 1][0], 0);
        __builtin_amdgcn_s_wait_tensorcnt(2);   // in-order => slice s pair has landed
      } else {
        __builtin_amdgcn_s_wait_tensorcnt(0);
      }
    }
    __syncthreads();   // slice s visible to all waves (orders xs staging on s==0 too)

    const float* mb = &msh[s & 1][mrow_off];
    const float* wb = &wsh[s & 1][wrow_off];
    const float* xb = &xs[bi * FIN + s * FS + 2 * h];

    // 16 K-steps: A = (w .* m) 16x4 tile, B = x broadcast across 16 columns.
    // C[o, n] += sum_k w[o,k]*m[b,o,k]*x[b,k]  for every column n (all equal).
#pragma unroll
    for (int k = 0; k < FS; k += 4) {
      v2f am = *(const v2f*)(mb + k);     // ds_load_b64, conflict-free (row stride 68)
      v2f aw = *(const v2f*)(wb + k);     // ds_load_b64
      v2f bx = *(const v2f*)(xb + k);     // ds_load_b64, broadcast within lane-halves
      v2f a;
      a.x = am.x * aw.x;
      a.y = am.y * aw.y;
      c_acc = __builtin_amdgcn_wmma_f32_16x16x4_f32(
          false, a, false, bx, (short)0, c_acc, false, false);
    }
    __syncthreads();   // done reading buffers before they are overwritten (slice s+2)
  }

  // C layout: vgpr p, lane l -> M = p + 8*h, N = lane&15; all columns identical.
  // Lanes with N==0 (lanes 0 and 16) publish their 8 rows.
  if (r == 0) {
    const int obase = o0 + g * 16 + 8 * h;
#pragma unroll
    for (int p = 0; p < 8; ++p) {
      out[(size_t)(b0 + bi) * FOUT + obase + p] = c_acc[p] + bias[obase + p];
    }
  }
}

extern "C" void kernel_launch(void* const* d_in, const int* in_sizes, int n_in,
                              void* d_out, int out_size, void* d_ws, size_t ws_size,
                              hipStream_t stream) {
  const float* x    = (const float*)d_in[0];
  const float* m    = (const float*)d_in[1];
  const float* w    = (const float*)d_in[2];
  const float* bias = (const float*)d_in[3];
  float* out = (float*)d_out;

  dim3 grid((BATCH / BT) * (FOUT / OTILE));   // 128 * 32 = 4096 blocks
  fused_dwa_wmma_kernel<<<grid, 128, 0, stream>>>(x, m, w, bias, out);
}